// BiEncoder_63024350101542
// MI455X (gfx1250) — compile-verified
//
#include <hip/hip_runtime.h>

// Problem constants from the reference
#define NB     64      // batch
#define SEQ    512     // sequence length
#define DIM    1024    // feature dim
#define MSTART 1
#define MEND   2

// Tiling: each block handles (batch b, one D-chunk of 512 floats).
// 128 threads/block (4 wave32), each lane owns 16 contiguous bytes of a row.
constexpr int DCH      = 2;             // D chunks
constexpr int CHUNK    = DIM / DCH;     // 512 floats per block
constexpr int TPB      = CHUNK / 4;     // 128 threads, 4 floats (16B) each
constexpr int DEPTH    = 8;             // async pipeline depth (ring of row buffers)
constexpr int ROWBYTES = CHUNK * 4;     // 2048 bytes per row-chunk

// ---- CDNA5 async global->LDS DMA (ASYNCcnt-tracked), via inline asm ----
// LDS destination offset comes from the vdst VGPR; GV mode: mem addr from 64-bit VGPR pair.
// s_wait_dscnt 0 first: guarantees the previous ds_load of this ring slot has
// fully read LDS before the async engine overwrites it (WAR).
__device__ __forceinline__ void async_row_load(uint32_t lds_off, const float* gptr) {
  asm volatile("s_wait_dscnt 0\n\t"
               "global_load_async_to_lds_b128 %0, %1, off"
               :: "v"(lds_off), "v"(gptr)
               : "memory");
}

// Wait until at most n async ops are outstanding (async loads complete in
// order, so asynccnt<=n proves the oldest issued-n rows have landed in LDS).
__device__ __forceinline__ void wait_async_le(int n) {
  switch (n) {
    case 0:  asm volatile("s_wait_asynccnt 0" ::: "memory"); break;
    case 1:  asm volatile("s_wait_asynccnt 1" ::: "memory"); break;
    case 2:  asm volatile("s_wait_asynccnt 2" ::: "memory"); break;
    case 3:  asm volatile("s_wait_asynccnt 3" ::: "memory"); break;
    case 4:  asm volatile("s_wait_asynccnt 4" ::: "memory"); break;
    case 5:  asm volatile("s_wait_asynccnt 5" ::: "memory"); break;
    case 6:  asm volatile("s_wait_asynccnt 6" ::: "memory"); break;
    default: asm volatile("s_wait_asynccnt 7" ::: "memory"); break;
  }
}

// ---- Kernel 1: per-batch first occurrence of MENTION_START / MENTION_END ----
__global__ void __launch_bounds__(256)
span_kernel(const int* __restrict__ ids, int* __restrict__ spans) {
  __shared__ int s[2];
  const int b = blockIdx.x;
  if (threadIdx.x < 2) s[threadIdx.x] = SEQ;   // "not found" == L, like reference
  __syncthreads();
  for (int p = threadIdx.x; p < SEQ; p += blockDim.x) {
    const int v = ids[b * SEQ + p];
    if (v == MSTART) atomicMin(&s[0], p);
    if (v == MEND)   atomicMin(&s[1], p);
  }
  __syncthreads();
  if (threadIdx.x == 0) {
    spans[2 * b + 0] = s[0];
    spans[2 * b + 1] = s[1];
  }
}

// ---- Kernel 2: masked span-max via async-pipelined row streaming ----
__global__ void __launch_bounds__(TPB)
span_max_kernel(const float* __restrict__ va, const int* __restrict__ spans,
                const float* __restrict__ mu, float* __restrict__ out) {
  __shared__ float buf[DEPTH][CHUNK];          // 8 x 2KB ring = 16KB LDS

  const int b   = blockIdx.x;
  const int col = blockIdx.y * CHUNK + threadIdx.x * 4;
  const float* rowBase = va + (size_t)b * SEQ * DIM + col;

  // cls = row 0 (direct coalesced b128 load; only needed once)
  const float4 cls = *(const float4*)rowBase;

  const int f1 = spans[2 * b + 0];
  const int f2 = spans[2 * b + 1];
  const int rs = f1 + 1;                       // first row with pos > first1
  const int re = (f2 < SEQ) ? f2 : SEQ;        // exclusive end: pos < first2

  float4 vec = cls;                            // fallback when span empty
  if (re > rs) {                               // uniform per block -> EXEC all-1s
    // low 32 bits of a flat LDS pointer == wave-relative LDS byte offset
    const uint32_t lds0 = (uint32_t)(uintptr_t)(&buf[0][0]) + threadIdx.x * 16u;
    const int nRows = re - rs;

    int issued = 0;
    for (; issued < DEPTH && issued < nRows; ++issued)
      async_row_load(lds0 + (uint32_t)issued * ROWBYTES,
                     rowBase + (size_t)(rs + issued) * DIM);

    float m0 = -INFINITY, m1 = -INFINITY, m2 = -INFINITY, m3 = -INFINITY;
    for (int i = 0; i < nRows; ++i) {
      wait_async_le(issued - i - 1);           // row i has landed in LDS
      const float4 v =
          *(const float4*)&buf[i & (DEPTH - 1)][threadIdx.x * 4];
      m0 = fmaxf(m0, v.x);
      m1 = fmaxf(m1, v.y);
      m2 = fmaxf(m2, v.z);
      m3 = fmaxf(m3, v.w);
      if (issued < nRows) {                    // refill the slot just consumed
        async_row_load(lds0 + (uint32_t)(issued & (DEPTH - 1)) * ROWBYTES,
                       rowBase + (size_t)(rs + issued) * DIM);
        ++issued;
      }
    }
    vec = make_float4(m0, m1, m2, m3);
  }

  const float m = mu[0];
  const float4 o = make_float4(cls.x + m * vec.x, cls.y + m * vec.y,
                               cls.z + m * vec.z, cls.w + m * vec.w);
  *(float4*)(out + (size_t)b * DIM + col) = o;
}

// ---- Host launcher ----
extern "C" void kernel_launch(void* const* d_in, const int* in_sizes, int n_in,
                              void* d_out, int out_size, void* d_ws, size_t ws_size,
                              hipStream_t stream) {
  const float* va  = (const float*)d_in[0];   // (B, L, D) f32
  const int*   ids = (const int*)d_in[1];     // (B, L) i32
  const float* mu  = (const float*)d_in[2];   // (1,) f32
  float*       out = (float*)d_out;           // (B, D) f32
  int*         spans = (int*)d_ws;            // 2 ints per batch (512 B)

  span_kernel<<<NB, 256, 0, stream>>>(ids, spans);
  span_max_kernel<<<dim3(NB, DCH), TPB, 0, stream>>>(va, spans, mu, out);
}